// GNN_4698694221926
// MI455X (gfx1250) — compile-verified
//
#include <hip/hip_runtime.h>
#include <hip/hip_bf16.h>

// ---------------------------------------------------------------------------
// GCN (3-layer) for MI455X / gfx1250. f32 WMMA (16x16x4) for the GEMMs,
// L2-resident atomic scatter for the graph aggregation.
//
// GEMM v2: Nc is a compile-time constant (immediate-offset addressing, no
// per-iteration 64-bit address math) and each wave computes the full 16 x Nc
// output strip (Nc/16 accumulator tiles) so A is streamed from HBM exactly
// once per layer (~51 MB instead of ~410 MB).
// ---------------------------------------------------------------------------

#define NNODES 100000
#define NEDGES 600000
#define NGRAPH 64
#define D_IN   128
#define D_HID  128
#define D_OUT  64

typedef float v2f __attribute__((ext_vector_type(2)));
typedef float v8f __attribute__((ext_vector_type(8)));

// ---------------- utility kernels ----------------

__global__ void gnn_zero_kernel(float* __restrict__ p, int n) {
    int i = blockIdx.x * blockDim.x + threadIdx.x;
    if (i < n) p[i] = 0.0f;
}

__global__ void gnn_deg_accum_kernel(const int* __restrict__ dst,
                                     const float* __restrict__ w,
                                     float* __restrict__ deg, int E) {
    int e = blockIdx.x * blockDim.x + threadIdx.x;
    if (e < E) atomicAdd(&deg[dst[e]], w[e]);
}

// deg += 1 (self loop), then overwrite with rsqrt(deg).
__global__ void gnn_deg_finalize_kernel(float* __restrict__ deg_dis, int n) {
    int i = blockIdx.x * blockDim.x + threadIdx.x;
    if (i < n) {
        float d = deg_dis[i] + 1.0f;
        deg_dis[i] = (d > 0.0f) ? rsqrtf(d) : 0.0f;
    }
}

__global__ void gnn_norm_kernel(const int* __restrict__ src,
                                const int* __restrict__ dst,
                                const float* __restrict__ w,
                                const float* __restrict__ dis,
                                float* __restrict__ norm, int E) {
    int e = blockIdx.x * blockDim.x + threadIdx.x;
    if (e < E) norm[e] = dis[src[e]] * w[e] * dis[dst[e]];
}

// ---------------- WMMA GEMM: C[M,Nc] = A[M,K] @ W[K,Nc] ----------------
// One wave32 per 16-row strip; the wave owns all Nc/16 column tiles.
//
// Fragment layouts per CDNA5 ISA 7.12.2:
//  A 16x4 f32 : lanes 0-15 -> M=lane, K-pair {0,1}; lanes 16-31 -> K-pair {2,3}
//  B 4x16 f32 : lanes 0-15 -> N=lane, K-pair {0,1}; lanes 16-31 -> K-pair {2,3}
//  C/D 16x16  : VGPR r -> M=r (lanes 0-15) / M=r+8 (lanes 16-31), N=lane&15
template <int K, int Nc>
__global__ __launch_bounds__(256)
void gnn_gemm_wmma_f32(const float* __restrict__ A,
                       const float* __restrict__ W,
                       float* __restrict__ C, int M) {
    constexpr int T = Nc / 16;                 // accumulator tiles per wave
    const int tileM = (blockIdx.x * blockDim.x + threadIdx.x) >> 5;
    const int lane  = threadIdx.x & 31;
    if (tileM * 16 >= M) return;               // wave-uniform exit (EXEC all-1s)

    const int l15     = lane & 15;
    const int halfsel = lane >> 4;             // 0: lanes 0-15, 1: lanes 16-31
    const int koff    = halfsel << 1;          // K sub-offset 0 or 2

    // Per-lane base pointers; all loop accesses are immediate offsets.
    const float* __restrict__ aptr = A + (size_t)(tileM * 16 + l15) * K + koff;
    const float* __restrict__ wptr = W + (size_t)koff * Nc + l15;

    v8f acc[T];
#pragma unroll
    for (int t = 0; t < T; ++t) acc[t] = v8f{};

#pragma unroll
    for (int k = 0; k < K; k += 4) {
        v2f a;
        a.x = aptr[k];                          // A[row][k+koff]
        a.y = aptr[k + 1];                      // A[row][k+koff+1]
#pragma unroll
        for (int t = 0; t < T; ++t) {
            v2f b;
            b.x = wptr[k * Nc + t * 16];        // W[k+koff][t*16 + l15]
            b.y = wptr[(k + 1) * Nc + t * 16];  // W[k+koff+1][t*16 + l15]
            acc[t] = __builtin_amdgcn_wmma_f32_16x16x4_f32(
                         /*neg_a=*/false, a, /*neg_b=*/false, b,
                         /*c_mod=*/(short)0, acc[t],
                         /*reuse_a=*/false, /*reuse_b=*/false);
        }
    }

    float* __restrict__ cptr = C + (size_t)(tileM * 16 + (halfsel << 3)) * Nc + l15;
#pragma unroll
    for (int t = 0; t < T; ++t) {
#pragma unroll
        for (int r = 0; r < 8; ++r) {
            cptr[r * Nc + t * 16] = acc[t][r];
        }
    }
}

// ---------------- edge scatter: agg[dst] += xw[src] * norm ----------------
// D/4 threads per edge; float4 gather (L2-resident), 4 f32 atomics.
template <int D>
__global__ void gnn_scatter_kernel(const float* __restrict__ xw,
                                   const int* __restrict__ src,
                                   const int* __restrict__ dst,
                                   const float* __restrict__ norm,
                                   float* __restrict__ agg, int E) {
    constexpr int CPE = D >> 2;               // float4 chunks per edge
    int t = blockIdx.x * blockDim.x + threadIdx.x;
    int e = t / CPE;
    if (e >= E) return;
    int c = (t - e * CPE) << 2;
    int s = src[e], d = dst[e];
    float nw = norm[e];
    const float4 v = *(const float4*)(xw + (size_t)s * D + c);
    float* o = agg + (size_t)d * D + c;
    atomicAdd(o + 0, v.x * nw);
    atomicAdd(o + 1, v.y * nw);
    atomicAdd(o + 2, v.z * nw);
    atomicAdd(o + 3, v.w * nw);
}

// ---------------- fused self-loop + bias + (optional) ReLU ----------------
// h[t] = act( agg[t] + xw[t] * dis[n]^2 + b[d] )   (agg and h may alias)
template <int D, int RELU>
__global__ void gnn_finalize_kernel(const float* __restrict__ agg,
                                    const float* __restrict__ xw,
                                    const float* __restrict__ dis,
                                    const float* __restrict__ bias,
                                    float* __restrict__ h, int n) {
    int t = blockIdx.x * blockDim.x + threadIdx.x;
    if (t >= n * D) return;
    int node = t / D;
    int d = t - node * D;
    float s = dis[node];
    float v = agg[t] + xw[t] * (s * s) + bias[d];
    if (RELU) v = fmaxf(v, 0.0f);
    h[t] = v;
}

// ---------------- pooling ----------------

__global__ void gnn_pool_accum_kernel(const float* __restrict__ h,
                                      const int* __restrict__ batch,
                                      float* __restrict__ psum, int n) {
    int t = blockIdx.x * blockDim.x + threadIdx.x;
    if (t >= n * D_OUT) return;
    int node = t >> 6;              // D_OUT == 64
    int d = t & 63;
    atomicAdd(&psum[(batch[node] << 6) + d], h[t]);
}

__global__ void gnn_pool_count_kernel(const int* __restrict__ batch,
                                      float* __restrict__ pcnt, int n) {
    int i = blockIdx.x * blockDim.x + threadIdx.x;
    if (i < n) atomicAdd(&pcnt[batch[i]], 1.0f);
}

__global__ void gnn_pool_div_kernel(const float* __restrict__ psum,
                                    const float* __restrict__ pcnt,
                                    float* __restrict__ out) {
    int t = blockIdx.x * blockDim.x + threadIdx.x;
    if (t < NGRAPH * D_OUT) out[t] = psum[t] / fmaxf(pcnt[t >> 6], 1.0f);
}

// ---------------- workspace layout (floats, 64B-aligned offsets) ----------

static constexpr size_t OFF_DIS  = 0;                            // N
static constexpr size_t OFF_NORM = 100352;                       // E
static constexpr size_t OFF_A    = OFF_NORM + 600064;            // N*128 (xw)
static constexpr size_t OFF_B    = OFF_A + (size_t)NNODES * 128; // N*128 (h)
static constexpr size_t OFF_PSUM = OFF_B + (size_t)NNODES * 128; // 64*64
static constexpr size_t OFF_PCNT = OFF_PSUM + NGRAPH * D_OUT;    // 64

extern "C" void kernel_launch(void* const* d_in, const int* in_sizes, int n_in,
                              void* d_out, int out_size, void* d_ws, size_t ws_size,
                              hipStream_t stream) {
    const float* x    = (const float*)d_in[0];
    const int*   eidx = (const int*)d_in[1];
    const int*   batch= (const int*)d_in[2];
    const float* ew   = (const float*)d_in[3];
    const float* W1   = (const float*)d_in[4];
    const float* b1   = (const float*)d_in[5];
    const float* W2   = (const float*)d_in[6];
    const float* b2   = (const float*)d_in[7];
    const float* W3   = (const float*)d_in[8];
    const float* b3   = (const float*)d_in[9];
    float* out = (float*)d_out;

    const int* src = eidx;           // edge_index[0]
    const int* dst = eidx + NEDGES;  // edge_index[1]

    float* ws   = (float*)d_ws;
    float* dis  = ws + OFF_DIS;
    float* norm = ws + OFF_NORM;
    float* bufA = ws + OFF_A;        // xw (GEMM output)
    float* bufB = ws + OFF_B;        // aggregated / layer output
    float* psum = ws + OFF_PSUM;
    float* pcnt = ws + OFF_PCNT;

    const int TB = 256;
    auto blocks = [](long long n, int tb) { return (int)((n + tb - 1) / tb); };

    // --- normalization coefficients ---
    gnn_zero_kernel<<<blocks(NNODES, TB), TB, 0, stream>>>(dis, NNODES);
    gnn_deg_accum_kernel<<<blocks(NEDGES, TB), TB, 0, stream>>>(dst, ew, dis, NEDGES);
    gnn_deg_finalize_kernel<<<blocks(NNODES, TB), TB, 0, stream>>>(dis, NNODES);
    gnn_norm_kernel<<<blocks(NEDGES, TB), TB, 0, stream>>>(src, dst, ew, dis, norm, NEDGES);

    const long long gemmThreads = (long long)(NNODES / 16) * 32;  // one wave / 16 rows

    // --- layer 1: h1 = relu(scatter(x@W1) + self + b1) ---
    gnn_gemm_wmma_f32<128, 128><<<blocks(gemmThreads, TB), TB, 0, stream>>>(x, W1, bufA, NNODES);
    gnn_zero_kernel<<<blocks((long long)NNODES * D_HID, TB), TB, 0, stream>>>(bufB, NNODES * D_HID);
    gnn_scatter_kernel<D_HID><<<blocks((long long)NEDGES * (D_HID / 4), TB), TB, 0, stream>>>(
        bufA, src, dst, norm, bufB, NEDGES);
    gnn_finalize_kernel<D_HID, 1><<<blocks((long long)NNODES * D_HID, TB), TB, 0, stream>>>(
        bufB, bufA, dis, b1, bufB, NNODES);

    // --- layer 2 ---
    gnn_gemm_wmma_f32<128, 128><<<blocks(gemmThreads, TB), TB, 0, stream>>>(bufB, W2, bufA, NNODES);
    gnn_zero_kernel<<<blocks((long long)NNODES * D_HID, TB), TB, 0, stream>>>(bufB, NNODES * D_HID);
    gnn_scatter_kernel<D_HID><<<blocks((long long)NEDGES * (D_HID / 4), TB), TB, 0, stream>>>(
        bufA, src, dst, norm, bufB, NEDGES);
    gnn_finalize_kernel<D_HID, 1><<<blocks((long long)NNODES * D_HID, TB), TB, 0, stream>>>(
        bufB, bufA, dis, b2, bufB, NNODES);

    // --- layer 3 (no relu) ---
    gnn_gemm_wmma_f32<128, 64><<<blocks(gemmThreads, TB), TB, 0, stream>>>(bufB, W3, bufA, NNODES);
    gnn_zero_kernel<<<blocks((long long)NNODES * D_OUT, TB), TB, 0, stream>>>(bufB, NNODES * D_OUT);
    gnn_scatter_kernel<D_OUT><<<blocks((long long)NEDGES * (D_OUT / 4), TB), TB, 0, stream>>>(
        bufA, src, dst, norm, bufB, NEDGES);
    gnn_finalize_kernel<D_OUT, 0><<<blocks((long long)NNODES * D_OUT, TB), TB, 0, stream>>>(
        bufB, bufA, dis, b3, bufB, NNODES);

    // --- global mean pool ---
    gnn_zero_kernel<<<blocks(NGRAPH * D_OUT + NGRAPH, TB), TB, 0, stream>>>(psum, NGRAPH * D_OUT + NGRAPH);
    gnn_pool_accum_kernel<<<blocks((long long)NNODES * D_OUT, TB), TB, 0, stream>>>(bufB, batch, psum, NNODES);
    gnn_pool_count_kernel<<<blocks(NNODES, TB), TB, 0, stream>>>(batch, pcnt, NNODES);
    gnn_pool_div_kernel<<<blocks(NGRAPH * D_OUT, TB), TB, 0, stream>>>(psum, pcnt, out);
}